// QuantumLayer_61340722921806
// MI455X (gfx1250) — compile-verified
//
#include <hip/hip_runtime.h>
#include <hip/hip_bf16.h>

typedef __attribute__((ext_vector_type(2))) float v2f;
typedef __attribute__((ext_vector_type(8))) float v8f;

#define NQ   8
#define DIM  256
#define PST  258   // padded LDS row stride (floats): 258*4 mod 256 = 8 -> conflict-free, 8B aligned

// Parity permutation of the K axis: even-parity j -> [0,128), odd-parity -> [128,256).
// jp = (j>>1) + 128*parity(j)  (bijection on 8-bit indices)

// ---------------------------------------------------------------------------
// Kernel 1: build the fixed 256x256 layer unitary from weights and store it as
// two real, parity-permuted matrices:
//   B1[i][jp] =  Re U[i][j]  (even parity j)   |  -Im U[i][j]  (odd parity j)
//   B2[i][jp] =  Im U[i][j]  (even parity j)   |   Re U[i][j]  (odd parity j)
// One block per basis column j; 128 threads simulate the 192-gate circuit in LDS.
// ---------------------------------------------------------------------------
__global__ __launch_bounds__(128) void build_unitary(
    const float* __restrict__ w,      // (8,8,2)
    float* __restrict__ B1, float* __restrict__ B2)
{
    __shared__ float sr[DIM], si[DIM];
    const int tid = threadIdx.x;      // 0..127
    const int col = blockIdx.x;       // basis column j

    for (int i = tid; i < DIM; i += 128) { sr[i] = 0.f; si[i] = 0.f; }
    __syncthreads();
    if (tid == 0) sr[col] = 1.f;
    __syncthreads();

    for (int l = 0; l < 8; ++l) {
        for (int q = 0; q < 8; ++q) {
            const int p   = 7 - q;
            const int i0  = ((tid >> p) << (p + 1)) | (tid & ((1 << p) - 1));
            const int i1  = i0 | (1 << p);
            {   // RY(w[l][q][0])
                const float th = 0.5f * w[(l * 8 + q) * 2 + 0];
                const float c = cosf(th), s = sinf(th);
                const float a0r = sr[i0], a0i = si[i0];
                const float a1r = sr[i1], a1i = si[i1];
                sr[i0] = c * a0r - s * a1r;  si[i0] = c * a0i - s * a1i;
                sr[i1] = s * a0r + c * a1r;  si[i1] = s * a0i + c * a1i;
            }
            __syncthreads();
            {   // RZ(w[l][q][1]): |0> *= (c - i s), |1> *= (c + i s)
                const float th = 0.5f * w[(l * 8 + q) * 2 + 1];
                const float c = cosf(th), s = sinf(th);
                const float a0r = sr[i0], a0i = si[i0];
                const float a1r = sr[i1], a1i = si[i1];
                sr[i0] = a0r * c + a0i * s;  si[i0] = a0i * c - a0r * s;
                sr[i1] = a1r * c - a1i * s;  si[i1] = a1i * c + a1r * s;
            }
            __syncthreads();
        }
        for (int q = 0; q < 8; ++q) {   // CNOT(q, (q+1)%8), wire0 = MSB
            const int pc_ = 7 - q;
            const int pt_ = 7 - ((q + 1) & 7);
            if (tid < 64) {
                const int pmin = pc_ < pt_ ? pc_ : pt_;
                const int pmax = pc_ < pt_ ? pt_ : pc_;
                int a = ((tid >> pmin) << (pmin + 1)) | (tid & ((1 << pmin) - 1));
                a     = ((a   >> pmax) << (pmax + 1)) | (a   & ((1 << pmax) - 1));
                const int i0 = a | (1 << pc_);      // control=1, target=0
                const int i1 = i0 | (1 << pt_);     // control=1, target=1
                const float tr = sr[i0], ti = si[i0];
                sr[i0] = sr[i1];  si[i0] = si[i1];
                sr[i1] = tr;      si[i1] = ti;
            }
            __syncthreads();
        }
    }

    const int par = __popc((unsigned)col) & 1;
    const int jp  = (col >> 1) + (par << 7);       // permuted column slot
    for (int i = tid; i < DIM; i += 128) {
        B1[i * DIM + jp] = par ? -si[i] : sr[i];
        B2[i * DIM + jp] = par ?  sr[i] : si[i];
    }
}

// ---------------------------------------------------------------------------
// Kernel 2: fused real GEMM pair (A x B1 -> Sr, A x B2 -> Si) + |.|^2 +
// signed Z reduction. Block = 16 batch rows, 8 waves; each wave owns two
// 16-wide state tiles. 2x v_wmma_f32_16x16x4_f32 per K-step.
// ---------------------------------------------------------------------------
__global__ __launch_bounds__(256) void qsim_wmma(
    const float* __restrict__ x,                          // (B, 8)
    const float* __restrict__ B1, const float* __restrict__ B2,
    float* __restrict__ out)                              // (B, 8)
{
    __shared__ float cs[16][8], sn[16][8];
    __shared__ float Ap[16 * PST];                        // parity-permuted real psi0 coeffs
    __shared__ float outs[16][8];

    const int tid = threadIdx.x;
    const int b0  = blockIdx.x * 16;

    // Phase 0: per-row cos/sin of half-angles; zero the output staging.
    if (tid < 128) {
        const int m = tid >> 3, q = tid & 7;
        const float t = 0.5f * x[(b0 + m) * 8 + q];
        cs[m][q] = cosf(t);
        sn[m][q] = sinf(t);
        outs[m][q] = 0.f;
    }
    __syncthreads();

    // Phase 1: build permuted product-state coefficients.
    // jp < 128: underlying j has even parity, psi0[j] = +-mag   (real)
    // jp >= 128: odd parity, psi0[j] = i*(+-mag); the +-mag goes into A.
    for (int e = tid; e < 16 * DIM; e += 256) {
        const int m  = e >> 8, jp = e & 255;
        const int t7 = jp & 127;
        const int hi = jp >> 7;                                   // parity region
        const int b  = (__popc((unsigned)t7) & 1) ^ hi;           // reconstructed LSB
        const int j  = (t7 << 1) | b;                             // underlying index
        float mag = 1.f;
#pragma unroll
        for (int q = 0; q < 8; ++q)
            mag *= ((j >> (7 - q)) & 1) ? sn[m][q] : cs[m][q];
        const int pc  = __popc((unsigned)j);
        const int neg = ((pc >> 1) & 1) ^ hi;                     // sign of (-i)^pc coeff
        Ap[m * PST + jp] = neg ? -mag : mag;
    }
    __syncthreads();

    // Phase 2: WMMA GEMMs + fused epilogue.
    const int wave = tid >> 5;
    const int lane = tid & 31;
    const int half = lane >> 4;       // 0: K=0,1 / M=v ; 1: K=2,3 / M=v+8
    const int mrow = lane & 15;       // A row (M) and B column (N) for this lane
    const int koff = half * 2;

    float acc[64];                    // acc[q*8+v]: signed prob partials
#pragma unroll
    for (int t = 0; t < 64; ++t) acc[t] = 0.f;

#pragma unroll
    for (int tt = 0; tt < 2; ++tt) {
        const int n0   = (wave * 2 + tt) * 16;
        const int icol = n0 + mrow;   // global state index handled by this lane
        v8f accR = {};
        v8f accI = {};
        const float2* b1p = reinterpret_cast<const float2*>(B1 + (size_t)icol * DIM);
        const float2* b2p = reinterpret_cast<const float2*>(B2 + (size_t)icol * DIM);
        const float*  apr = Ap + mrow * PST;

        for (int k0 = 0; k0 < DIM; k0 += 4) {
            const int ka = k0 + koff;
            v2f a;  a.x = apr[ka];  a.y = apr[ka + 1];
            const float2 t1 = b1p[ka >> 1]; v2f b1; b1.x = t1.x; b1.y = t1.y;
            const float2 t2 = b2p[ka >> 1]; v2f b2; b2.x = t2.x; b2.y = t2.y;
            accR = __builtin_amdgcn_wmma_f32_16x16x4_f32(false, a, false, b1, (short)0, accR, false, false);
            accI = __builtin_amdgcn_wmma_f32_16x16x4_f32(false, a, false, b2, (short)0, accI, false, false);
        }

        float sg[8];
#pragma unroll
        for (int q = 0; q < 8; ++q)
            sg[q] = 1.f - 2.f * (float)((icol >> (7 - q)) & 1);
#pragma unroll
        for (int v = 0; v < 8; ++v) {
            const float p = accR[v] * accR[v] + accI[v] * accI[v];
#pragma unroll
            for (int q = 0; q < 8; ++q)
                acc[q * 8 + v] = fmaf(p, sg[q], acc[q * 8 + v]);
        }
    }

    // Reduce across the 16 lanes of each half (N dimension of the tile).
#pragma unroll
    for (int off = 8; off >= 1; off >>= 1) {
#pragma unroll
        for (int t = 0; t < 64; ++t)
            acc[t] += __shfl_xor(acc[t], off, 32);
    }
    if (mrow == 0) {
#pragma unroll
        for (int v = 0; v < 8; ++v) {
            const int m = v + half * 8;   // C/D layout: lanes>=16 hold M=v+8
#pragma unroll
            for (int q = 0; q < 8; ++q)
                atomicAdd(&outs[m][q], acc[q * 8 + v]);
        }
    }
    __syncthreads();

    if (tid < 128) {
        const int m = tid >> 3, q = tid & 7;
        out[(b0 + m) * 8 + q] = outs[m][q];
    }
}

// ---------------------------------------------------------------------------
extern "C" void kernel_launch(void* const* d_in, const int* in_sizes, int n_in,
                              void* d_out, int out_size, void* d_ws, size_t ws_size,
                              hipStream_t stream) {
    const float* x = (const float*)d_in[0];   // (65536, 8)
    const float* w = (const float*)d_in[1];   // (8, 8, 2)
    float* out = (float*)d_out;               // (65536, 8)

    float* B1 = (float*)d_ws;                 // 256*256 floats
    float* B2 = B1 + DIM * DIM;               // total 512 KB of workspace

    const int batch = in_sizes[0] / NQ;       // 65536

    build_unitary<<<DIM, 128, 0, stream>>>(w, B1, B2);
    qsim_wmma<<<batch / 16, 256, 0, stream>>>(x, B1, B2, out);
}